// AngularCosDiff_29824252903586
// MI455X (gfx1250) — compile-verified
//
#include <hip/hip_runtime.h>

// ANI-1x angular term, CDNA5 (gfx1250, wave32).
//
// Roofline: 24B in + 144B out per pair -> ~704MB total -> ~30us @ 23.3TB/s:
// memory bound, store-dominated. The 36-wide quadratic-form exponent is a
// (Px4)x(4x36) GEMM: Y = [cos^2, cos, davg^2, davg] * W + bias(col), done with
// V_WMMA_F32_16X16X4_F32 (K=4 exact; column bias rides in the C accumulator).
// W/bias pre-scaled by log2(e) -> elementwise tail is bare v_exp_f32 + v_mul,
// then non-temporal b32 stores at immediate offsets from one base per M-tile.
//
// The main kernel is completely guard-free (covers floor(P/256)*256 rows with
// whole waves, EXEC all-1s except the compile-time n<4 mask on the last
// N-tile); leftover rows go to a tiny scalar tail kernel.

typedef __attribute__((ext_vector_type(2))) float v2f;
typedef __attribute__((ext_vector_type(8))) float v8f;

__global__ __launch_bounds__(256)
void ani_angular_wmma_main(const float* __restrict__ vec12,  // (2, P, 3)
                           const float* __restrict__ EtaA,   // (1,)
                           const float* __restrict__ ShfA,   // (4,)
                           const float* __restrict__ Gamma,  // (9,)
                           const float* __restrict__ ShfZ,   // (9,)
                           float* __restrict__ out,          // (P, 36)
                           int P)                             // for v1 offset
{
    const int lane = threadIdx.x & 31;
    const int wave = threadIdx.x >> 5;
    const int rowBase = (blockIdx.x * 8 + wave) * 32;   // always fully in-range

    const int n  = lane & 15;            // N (column) within a 16-wide tile
    const int kh = lane >> 4;            // 0: lo half (K=0,1), 1: hi half (K=2,3)

    // ---------------- B matrices (4x16 per N-tile) + column bias -------------
    // Y*log2e = cos^2*(-G*L) + cos*(2G*cz*L) + davg^2*(-eta*L) + davg*(2eta*sa*L)
    //           - (G*cz^2 + eta*sa^2)*L,   L = log2(e),  j = a*9 + z  (E==1)
    // B mirrors A: lane holds column N=lane%16; VGPR0=K even, VGPR1=K odd;
    // lo-half lanes K=0,1; hi-half lanes K=2,3.
    const float L2E = 1.44269504088896340736f;
    const float eta = EtaA[0];
    v2f   Bm[3];
    float bias[3];
#pragma unroll
    for (int t = 0; t < 3; ++t) {
        int j = t * 16 + n;
        v2f b; float bi;
        if (j < 36) {
            int a = j / 9, z = j % 9;
            float G  = Gamma[z];
            float cz = __cosf(ShfZ[z]);
            float sa = ShfA[a];
            if (kh == 0) { b.x = -G * L2E;   b.y = 2.0f * G * cz * L2E; }
            else         { b.x = -eta * L2E; b.y = 2.0f * eta * sa * L2E; }
            bi = -(G * cz * cz + eta * sa * sa) * L2E;
        } else {
            b.x = 0.0f; b.y = 0.0f; bi = 0.0f;
        }
        Bm[t] = b; bias[t] = bi;
    }

    // ---------------- per-lane row scalars (one pair per lane) ---------------
    int p = rowBase + lane;
    const float* v0p = vec12 + (size_t)p * 3;                   // -> b96 load
    const float* v1p = vec12 + (size_t)P * 3 + (size_t)p * 3;   // -> b96 load
    float x0 = v0p[0], y0 = v0p[1], z0 = v0p[2];
    float x1 = v1p[0], y1 = v1p[1], z1 = v1p[2];

    float d0sq = x0*x0 + y0*y0 + z0*z0;
    float d1sq = x1*x1 + y1*y1 + z1*z1;
    float dot  = x0*x1 + y0*y1 + z0*z1;
    float d0 = sqrtf(d0sq), d1 = sqrtf(d1sq);
    float cosv = dot / fmaxf(d0 * d1, 1e-10f);
    float davg = 0.5f * (d0 + d1);
    const float c2 = 3.5f * 3.5f;
    float f0 = c2 - d0sq, f1 = c2 - d1sq;
    float s  = 4.0f * (f0 * f0) * (f1 * f1) * (1.0f / (c2 * c2 * c2 * c2));

    // ---------------- 2 M-tiles of 16 rows, 3 N-tiles of 16 cols -------------
#pragma unroll
    for (int t = 0; t < 2; ++t) {
        // A (16x4 f32): lane L -> M=L%16; value = cos (lo half) / davg (hi
        // half); VGPR0 = value^2 (K even), VGPR1 = value (K odd).
        float a1 = (kh == 0) ? __shfl(cosv, t * 16 + n, 32)
                             : __shfl(davg, t * 16 + n, 32);
        v2f A; A.x = a1 * a1; A.y = a1;

        // scale 4*fc for the 8 D-rows this lane holds: M = v + 8*kh
        float srow[8];
#pragma unroll
        for (int v = 0; v < 8; ++v)
            srow[v] = __shfl(s, t * 16 + v + 8 * kh, 32);

        // one 64-bit store base per M-tile; everything else is imm offsets
        float* bt = out + (size_t)(rowBase + t * 16 + 8 * kh) * 36 + n;

#pragma unroll
        for (int nt = 0; nt < 3; ++nt) {
            v8f c;
#pragma unroll
            for (int v = 0; v < 8; ++v) c[v] = bias[nt];

            // D = A(16x4) * B(4x16) + C   -> v_wmma_f32_16x16x4_f32
            v8f d = __builtin_amdgcn_wmma_f32_16x16x4_f32(
                false, A, false, Bm[nt], (short)0, c, false, false);

            // N-tiles 0,1 fully in range (j<=31); tile 2 keeps cols n<4 (j<36)
            if (nt < 2 || n < 4) {
#pragma unroll
                for (int v = 0; v < 8; ++v)
                    __builtin_nontemporal_store(
                        srow[v] * __builtin_amdgcn_exp2f(d[v]),
                        bt + nt * 16 + v * 36);
            }
        }
    }
}

// Scalar tail: one thread per output element for rows [pStart, P).
__global__ void ani_angular_tail(const float* __restrict__ vec12,
                                 const float* __restrict__ EtaA,
                                 const float* __restrict__ ShfA,
                                 const float* __restrict__ Gamma,
                                 const float* __restrict__ ShfZ,
                                 float* __restrict__ out,
                                 int P, int pStart)
{
    int idx = blockIdx.x * blockDim.x + threadIdx.x;   // element in tail region
    int p = pStart + idx / 36;
    int j = idx % 36;
    if (p >= P) return;
    int a = j / 9, z = j % 9;

    const float* v0p = vec12 + (size_t)p * 3;
    const float* v1p = vec12 + (size_t)P * 3 + (size_t)p * 3;
    float x0 = v0p[0], y0 = v0p[1], z0 = v0p[2];
    float x1 = v1p[0], y1 = v1p[1], z1 = v1p[2];
    float d0sq = x0*x0 + y0*y0 + z0*z0;
    float d1sq = x1*x1 + y1*y1 + z1*z1;
    float dot  = x0*x1 + y0*y1 + z0*z1;
    float d0 = sqrtf(d0sq), d1 = sqrtf(d1sq);
    float cosv = dot / fmaxf(d0 * d1, 1e-10f);
    float davg = 0.5f * (d0 + d1);
    const float c2 = 3.5f * 3.5f;
    float f0 = c2 - d0sq, f1 = c2 - d1sq;
    float fc = (f0 * f0) * (f1 * f1) * (1.0f / (c2 * c2 * c2 * c2));

    float dc = cosv - __cosf(ShfZ[z]);
    float dd = davg - ShfA[a];
    float y  = Gamma[z] * dc * dc + EtaA[0] * dd * dd;
    out[(size_t)p * 36 + j] = 4.0f * fc * __expf(-y);
}

extern "C" void kernel_launch(void* const* d_in, const int* in_sizes, int n_in,
                              void* d_out, int out_size, void* d_ws, size_t ws_size,
                              hipStream_t stream) {
    const float* vec12 = (const float*)d_in[0];  // (2, P, 3) f32
    const float* EtaA  = (const float*)d_in[1];  // (1,)
    const float* ShfA  = (const float*)d_in[2];  // (4,)
    const float* Gamma = (const float*)d_in[3];  // (9,)
    const float* ShfZ  = (const float*)d_in[4];  // (9,)
    float* out = (float*)d_out;                  // (P, 36) f32

    int P = in_sizes[0] / 6;                     // 2*P*3 elements
    int fullBlocks = P / 256;                    // 8 waves x 32 rows per block
    if (fullBlocks > 0)
        ani_angular_wmma_main<<<fullBlocks, 256, 0, stream>>>(
            vec12, EtaA, ShfA, Gamma, ShfZ, out, P);

    int pStart = fullBlocks * 256;
    int tailElems = (P - pStart) * 36;           // < 256*36
    if (tailElems > 0)
        ani_angular_tail<<<(tailElems + 255) / 256, 256, 0, stream>>>(
            vec12, EtaA, ShfA, Gamma, ShfZ, out, P, pStart);
}